// CoAttention_TextImage_13391708029219
// MI455X (gfx1250) — compile-verified
//
#include <hip/hip_runtime.h>
#include <hip/hip_bf16.h>

// Problem dims (compile-time)
#define BB 8
#define SS 128
#define RR 49
#define HH 768

typedef float v2f __attribute__((ext_vector_type(2)));
typedef float v8f __attribute__((ext_vector_type(8)));

// ---------------------------------------------------------------------------
// Fused GEMM + tanh-dot epilogue body:
//   Y = A(MxK) @ W(KxN)          (K = N = 768, fp32 WMMA 16x16x4)
//   score[m] = sum_n tanh(Y[m,n] + bias[n]) * wv[n]
// One block (256 threads = 8 waves) owns a 16-row strip across all 768 cols.
// Each wave: 6 column tiles (16x96), A fragment reused across the 6 WMMAs.
// Epilogue: intra-half shfl_xor butterfly (wave32) + LDS cross-wave reduce,
// single deterministic store per row (no atomics, no zero-init required).
// ---------------------------------------------------------------------------
__device__ __forceinline__
void gemm_tanh_score_body(const float* __restrict__ A, const float* __restrict__ W,
                          const float* __restrict__ bias, const float* __restrict__ wv,
                          float* __restrict__ score, int M, int rowblk)
{
    constexpr int K = HH;
    constexpr int N = HH;
    const int lane = threadIdx.x & 31;
    const int wave = threadIdx.x >> 5;
    const int half = lane >> 4;   // 0: K pair {0,1}, 1: K pair {2,3}
    const int lm   = lane & 15;
    const int row0 = rowblk * 16;

    // A fragment address: row = row0+lm (clamped for ragged M), k = 2*half + j
    int arow = row0 + lm;
    if (arow > M - 1) arow = M - 1;
    const float* Ap = A + (size_t)arow * K + 2 * half;

    // This wave's first column of its 96-wide strip
    const int colw = wave * 96;
    const float* Wp = W + (size_t)(2 * half) * N + colw + lm;

    v8f acc[6];
#pragma unroll
    for (int j = 0; j < 6; ++j) acc[j] = (v8f){};

#pragma unroll 2
    for (int k = 0; k < K; k += 4) {
        v2f a = *(const v2f*)(Ap + k);               // global_load_b64
        const float* wk = Wp + (size_t)k * N;
#pragma unroll
        for (int j = 0; j < 6; ++j) {
            v2f b;
            b.x = wk[j * 16];                         // row k+2*half
            b.y = wk[j * 16 + N];                     // row k+2*half+1
            acc[j] = __builtin_amdgcn_wmma_f32_16x16x4_f32(
                false, a, false, b, (short)0, acc[j], false, false);
        }
    }

    // Epilogue: t[m] = sum_n tanh(c + bias[n]) * wv[n], reduced over this
    // wave's 96 columns. C layout: VGPR v <-> row (half*8+v), col lm.
    float racc[8];
#pragma unroll
    for (int v = 0; v < 8; ++v) racc[v] = 0.f;

#pragma unroll
    for (int j = 0; j < 6; ++j) {
        const int gn = colw + j * 16 + lm;
        const float bn = bias ? bias[gn] : 0.f;
        const float wn = wv[gn];
#pragma unroll
        for (int v = 0; v < 8; ++v) {
            float t = tanhf(acc[j][v] + bn) * wn;
            // reduce across the 16 lanes of this half (masks < 16 keep halves)
            t += __shfl_xor(t, 8, 32);
            t += __shfl_xor(t, 4, 32);
            t += __shfl_xor(t, 2, 32);
            t += __shfl_xor(t, 1, 32);
            racc[v] += t;
        }
    }

    __shared__ float sred[8][16];
    if (lm == 0) {
#pragma unroll
        for (int v = 0; v < 8; ++v) sred[wave][half * 8 + v] = racc[v];
    }
    __syncthreads();
    if (threadIdx.x < 16) {
        const int m = row0 + threadIdx.x;
        if (m < M) {
            float s = 0.f;
#pragma unroll
            for (int w = 0; w < 8; ++w) s += sred[w][threadIdx.x];
            score[m] = s;
        }
    }
}

// Single-job GEMM (stage 2 query-side projection)
__global__ __launch_bounds__(256)
void gemm_tanh_score(const float* __restrict__ A, const float* __restrict__ W,
                     const float* __restrict__ bias, const float* __restrict__ wv,
                     float* __restrict__ score, int M)
{
    gemm_tanh_score_body(A, W, bias, wv, score, M, blockIdx.x);
}

// Three independent GEMM jobs fused into one launch (st, si, sk are
// mutually independent): blocks [0,64) -> job0, [64,89) -> job1,
// [89,153) -> job2. Raises occupancy ~2.4x and drops two launch latencies.
struct GemmJob {
    const float* A;
    const float* W;
    const float* bias;
    const float* wv;
    float*       score;
    int          M;
};

__global__ __launch_bounds__(256)
void gemm_tanh_score_batch3(GemmJob j0, GemmJob j1, GemmJob j2, int n0, int n01)
{
    const int bx = blockIdx.x;
    if (bx < n0) {
        gemm_tanh_score_body(j0.A, j0.W, j0.bias, j0.wv, j0.score, j0.M, bx);
    } else if (bx < n01) {
        gemm_tanh_score_body(j1.A, j1.W, j1.bias, j1.wv, j1.score, j1.M, bx - n0);
    } else {
        gemm_tanh_score_body(j2.A, j2.W, j2.bias, j2.wv, j2.score, j2.M, bx - n01);
    }
}

// ---------------------------------------------------------------------------
// Stage 1: per (b,s): softmax over R=49 image regions, then
//   att_img[b,s,:] = sum_r p[r] * img[b,r,:]
// ---------------------------------------------------------------------------
__global__ __launch_bounds__(256)
void softmax_attend_img(const float* __restrict__ st, const float* __restrict__ si,
                        const float* __restrict__ ba1, const float* __restrict__ img,
                        float* __restrict__ out)
{
    const int bs = blockIdx.x;       // b*S + s
    const int b  = bs / SS;
    const int tid = threadIdx.x;

    __shared__ float p[RR];
    __shared__ float inv_s;

    if (tid < RR) p[tid] = st[bs] + si[b * RR + tid] + ba1[0];
    __syncthreads();
    if (tid == 0) {
        float mx = p[0];
        for (int r = 1; r < RR; ++r) mx = fmaxf(mx, p[r]);
        float sum = 0.f;
        for (int r = 0; r < RR; ++r) {
            float e = __expf(p[r] - mx);
            p[r] = e;
            sum += e;
        }
        inv_s = 1.0f / sum;
    }
    __syncthreads();
    const float inv = inv_s;
    const float* imgb = img + (size_t)b * RR * HH;

    for (int h = tid; h < HH; h += 256) {
        float a0 = 0.f;
#pragma unroll 7
        for (int r = 0; r < RR; ++r)
            a0 = fmaf(p[r], imgb[(size_t)r * HH + h], a0);
        out[(size_t)bs * HH + h] = a0 * inv;
    }
}

// ---------------------------------------------------------------------------
// Stage 2: per (b,s): softmax over S=128 text tokens, then
//   att_text[b,s,:] = sum_t p[t] * text[b,t,:]
// ---------------------------------------------------------------------------
__global__ __launch_bounds__(256)
void softmax_attend_text(const float* __restrict__ sq, const float* __restrict__ sk,
                         const float* __restrict__ ba2, const float* __restrict__ text,
                         float* __restrict__ out)
{
    const int bs = blockIdx.x;       // b*S + s
    const int b  = bs / SS;
    const int tid = threadIdx.x;

    __shared__ float p[SS];
    __shared__ float inv_s;

    if (tid < SS) p[tid] = sq[bs] + sk[b * SS + tid] + ba2[0];
    __syncthreads();
    if (tid == 0) {
        float mx = p[0];
        for (int t = 1; t < SS; ++t) mx = fmaxf(mx, p[t]);
        float sum = 0.f;
        for (int t = 0; t < SS; ++t) {
            float e = __expf(p[t] - mx);
            p[t] = e;
            sum += e;
        }
        inv_s = 1.0f / sum;
    }
    __syncthreads();
    const float inv = inv_s;
    const float* txb = text + (size_t)b * SS * HH;

    for (int h = tid; h < HH; h += 256) {
        float a0 = 0.f;
#pragma unroll 8
        for (int t = 0; t < SS; ++t)
            a0 = fmaf(p[t], txb[(size_t)t * HH + h], a0);
        out[(size_t)bs * HH + h] = a0 * inv;
    }
}

// ---------------------------------------------------------------------------
// Launch: 4 stream-ordered kernels (graph-capture safe, no atomics,
// all buffers fully overwritten every call -> deterministic).
// ws float layout: st[1024] | si[392] | sk[1024] | sq[1024]  (~14 KB)
// ---------------------------------------------------------------------------
extern "C" void kernel_launch(void* const* d_in, const int* in_sizes, int n_in,
                              void* d_out, int out_size, void* d_ws, size_t ws_size,
                              hipStream_t stream) {
    (void)in_sizes; (void)n_in; (void)out_size; (void)ws_size;

    const float* text = (const float*)d_in[0];   // (B,S,H)
    const float* img  = (const float*)d_in[1];   // (B,R,H)
    const float* W_t1 = (const float*)d_in[2];   // (H,H)
    const float* b_t1 = (const float*)d_in[3];   // (H,)
    const float* W_i1 = (const float*)d_in[4];   // (H,H)
    const float* w_a1 = (const float*)d_in[5];   // (2H,) [text | img]
    const float* b_a1 = (const float*)d_in[6];   // scalar
    const float* W_t2 = (const float*)d_in[7];   // (H,H)
    const float* W_i2 = (const float*)d_in[8];   // (H,H)
    const float* b_i2 = (const float*)d_in[9];   // (H,)
    const float* w_a2 = (const float*)d_in[10];  // (2H,) [img | text]
    const float* b_a2 = (const float*)d_in[11];  // scalar

    float* att_text = (float*)d_out;                         // first output
    float* att_img  = (float*)d_out + (size_t)BB * SS * HH;  // second output

    float* ws = (float*)d_ws;
    float* st = ws;                       // (B*S,)
    float* si = st + BB * SS;             // (B*R,)
    float* sk = si + BB * RR;             // (B*S,)
    float* sq = sk + BB * SS;             // (B*S,)

    const int M1 = BB * SS;   // 1024
    const int Mi = BB * RR;   // 392
    const int nb1 = (M1 + 15) / 16;  // 64 row-blocks
    const int nbi = (Mi + 15) / 16;  // 25 row-blocks
    dim3 blk(256);

    // stage 1 scores + stage 2 key-side scores: three independent GEMMs,
    // one launch (153 blocks)
    GemmJob j0 = { text, W_t1, b_t1,    w_a1,      st, M1 };
    GemmJob j1 = { img,  W_i1, nullptr, w_a1 + HH, si, Mi };
    GemmJob j2 = { text, W_t2, nullptr, w_a2 + HH, sk, M1 };
    gemm_tanh_score_batch3<<<dim3(nb1 + nbi + nb1), blk, 0, stream>>>(
        j0, j1, j2, nb1, nb1 + nbi);

    // stage 1 softmax + attended image features (second half of d_out)
    softmax_attend_img<<<BB * SS, blk, 0, stream>>>(st, si, b_a1, img, att_img);

    // stage 2 query-side scores from attended image features
    gemm_tanh_score<<<dim3(nb1), blk, 0, stream>>>(att_img, W_i2, b_i2, w_a2, sq, M1);

    // stage 2 softmax + attended text features (first half of d_out)
    softmax_attend_text<<<BB * SS, blk, 0, stream>>>(sq, sk, b_a2, text, att_text);
}